// VGAE_6055903887706
// MI455X (gfx1250) — compile-verified
//
#include <hip/hip_runtime.h>
#include <math.h>

#define N_NODES 12288
#define BATCH   8
#define NGRP    1536
#define E_EDGES 49152
#define T_FEAT  9
#define R_REL   4
#define H_DIM   128
#define K_PAD   16

typedef float v2f __attribute__((ext_vector_type(2)));
typedef float v8f __attribute__((ext_vector_type(8)));

// ---------------------------------------------------------------------------
// C (Nrows x 128) = A (Nrows x K) @ B (K x 128) [+ bias broadcast over rows]
// K is compile-time (16 or 128, multiple of 16) -> guard-free inner loop.
// One wave32 computes one 16x16 tile with V_WMMA_F32_16X16X4_F32.
// Block = 256 threads = 8 waves -> 16 rows x 128 cols per block.
// B is staged through LDS (16 K-rows = 8KB/chunk), shared by all 8 waves.
// A fragment (16x4 f32): VGPR0 lanes0-15 K=k0, lanes16-31 K=k0+2; VGPR1 K=k0+1/k0+3.
// B fragment (4x16 f32): VGPR0 lanes0-15 row k0, lanes16-31 row k0+2; VGPR1 k0+1/k0+3.
// C/D (16x16 f32): VGPR i: lanes0-15 -> M=i, lanes16-31 -> M=i+8, N = lane%16.
// ---------------------------------------------------------------------------
template<int K>
__global__ void __launch_bounds__(256)
gemm_n128_wmma(const float* __restrict__ A, const float* __restrict__ Bm,
               const float* __restrict__ bias, int has_bias, float* __restrict__ C)
{
    __shared__ float Bs[16 * H_DIM];              // 8 KB
    const int tid  = threadIdx.x;
    const int wave = tid >> 5;
    const int lane = tid & 31;
    const int row0 = blockIdx.x << 4;
    const int col0 = wave << 4;
    const int ml   = lane & 15;
    const int hi   = lane >> 4;                   // 0: lanes 0-15, 1: lanes 16-31

    const float* Arow = A + (size_t)(row0 + ml) * K;
    v8f acc = {};

    for (int k0 = 0; k0 < K; k0 += 16) {
        // cooperative stage of B[k0:k0+16, :] into LDS (fully coalesced b128)
        {
            const int kr = tid >> 5;              // 0..7
            const int c4 = (tid & 31) << 2;       // 0,4,...,124
            *(float4*)(Bs + kr * H_DIM + c4) =
                *(const float4*)(Bm + (size_t)(k0 + kr) * H_DIM + c4);
            *(float4*)(Bs + (kr + 8) * H_DIM + c4) =
                *(const float4*)(Bm + (size_t)(k0 + kr + 8) * H_DIM + c4);
        }
        __syncthreads();
#pragma unroll
        for (int kk = 0; kk < 16; kk += 4) {
            const int ka = kk + (hi << 1);
            v2f a = *(const v2f*)(Arow + k0 + ka);         // aligned 8B load
            v2f b;
            b.x = Bs[ka * H_DIM + col0 + ml];
            b.y = Bs[(ka + 1) * H_DIM + col0 + ml];
            acc = __builtin_amdgcn_wmma_f32_16x16x4_f32(
                false, a, false, b, (short)0, acc, false, false);
        }
        __syncthreads();
    }

    const float bv = has_bias ? bias[col0 + ml] : 0.0f;
    float* Cp = C + (size_t)(row0 + (hi << 3)) * H_DIM + col0 + ml;
#pragma unroll
    for (int i = 0; i < 8; ++i)
        Cp[(size_t)i * H_DIM] = acc[i] + bv;
}

// Pad (nrows x srccols) -> (nrows x 16), zero-fill cols >= srccols
__global__ void pad_rows16(const float* __restrict__ src, float* __restrict__ dst,
                           int nrows, int srccols)
{
    int i = blockIdx.x * blockDim.x + threadIdx.x;
    if (i >= nrows * 16) return;
    int c = i & 15, r = i >> 4;
    dst[i] = (c < srccols) ? src[r * srccols + c] : 0.0f;
}

// Pad (nmats x 9 x 128) -> (nmats x 16 x 128), zero-fill K-rows >= 9
__global__ void pad_w16(const float* __restrict__ src, float* __restrict__ dst, int nmats)
{
    int i = blockIdx.x * blockDim.x + threadIdx.x;
    if (i >= nmats * K_PAD * H_DIM) return;
    int h = i % H_DIM;
    int k = (i / H_DIM) & 15;
    int m = i / (K_PAD * H_DIM);
    dst[i] = (k < T_FEAT) ? src[(m * T_FEAT + k) * H_DIM + h] : 0.0f;
}

// One wave per edge: out[dst] += hr[src]  (128 floats = 4 per lane, f32 atomics)
__global__ void __launch_bounds__(256)
scatter_add_rel(const float* __restrict__ hr, float* __restrict__ outp,
                const int* __restrict__ esrc, const int* __restrict__ edst,
                const int* __restrict__ ety, int rel)
{
    const int wid  = (int)((blockIdx.x * blockDim.x + threadIdx.x) >> 5);
    const int lane = threadIdx.x & 31;
    if (wid >= E_EDGES) return;
    if (ety[wid] != rel) return;
    const int src = esrc[wid];
    const int dst = edst[wid];
    const float4 v = *(const float4*)(hr + (size_t)src * H_DIM + (lane << 2));
    float* o = outp + (size_t)dst * H_DIM + (lane << 2);
    atomicAdd(o + 0, v.x);
    atomicAdd(o + 1, v.y);
    atomicAdd(o + 2, v.z);
    atomicAdd(o + 3, v.w);
}

__global__ void relu_inplace(float* __restrict__ x, int n)
{
    int i = blockIdx.x * blockDim.x + threadIdx.x;
    if (i < n) { float v = x[i]; x[i] = v > 0.0f ? v : 0.0f; }
}

__global__ void z_combine(const float* __restrict__ mu, const float* __restrict__ ls,
                          const float* __restrict__ eps, float* __restrict__ z, int n)
{
    int i = blockIdx.x * blockDim.x + threadIdx.x;
    if (i < n) z[i] = mu[i] + eps[i] * expf(ls[i]);
}

// nt[n,t] = z[n,:] @ Wnt[:,t] + bnt[t]   (H=128, T=9; trivial GEMV)
__global__ void nt_gemv(const float* __restrict__ z, const float* __restrict__ Wnt,
                        const float* __restrict__ bnt, float* __restrict__ nt)
{
    int i = blockIdx.x * blockDim.x + threadIdx.x;
    if (i >= N_NODES * T_FEAT) return;
    int t = i % T_FEAT, n = i / T_FEAT;
    const float* zr = z + (size_t)n * H_DIM;
    float s = bnt[t];
    for (int k = 0; k < H_DIM; ++k) s += zr[k] * Wnt[k * T_FEAT + t];
    nt[i] = s;
}

// gz[b,h] = mean over NGRP nodes of z
__global__ void gz_mean(const float* __restrict__ z, float* __restrict__ gz)
{
    int i = blockIdx.x * blockDim.x + threadIdx.x;   // i = b*H + h
    if (i >= BATCH * H_DIM) return;
    int h = i % H_DIM, b = i / H_DIM;
    const float* zp = z + ((size_t)b * NGRP) * H_DIM + h;
    float s = 0.0f;
    for (int j = 0; j < NGRP; ++j) s += zp[(size_t)j * H_DIM];
    gz[i] = s * (1.0f / NGRP);
}

// Per-node scores: su, sv, Au[4], Av[4]
__global__ void node_scores(const float* __restrict__ z, const float* __restrict__ nt,
                            const float* __restrict__ we, const float* __restrict__ Wt,
                            float* __restrict__ su, float* __restrict__ sv,
                            float* __restrict__ Au, float* __restrict__ Av)
{
    int n = blockIdx.x * blockDim.x + threadIdx.x;
    if (n >= N_NODES) return;
    const float* zr = z + (size_t)n * H_DIM;
    const float* nr = nt + (size_t)n * T_FEAT;
    float s_u = 0.0f, s_v = 0.0f;
    float au[R_REL] = {0, 0, 0, 0}, av[R_REL] = {0, 0, 0, 0};
    for (int k = 0; k < H_DIM; ++k) {
        float zk = zr[k];
        s_u += zk * we[k];                            // wu_z = we[0:H]
        s_v += zk * we[H_DIM + T_FEAT + k];           // wv_z = we[H+T:2H+T]
        const float* wtu = Wt + (size_t)k * R_REL;
        const float* wtv = Wt + (size_t)(H_DIM + T_FEAT + k) * R_REL;
#pragma unroll
        for (int r = 0; r < R_REL; ++r) { au[r] += zk * wtu[r]; av[r] += zk * wtv[r]; }
    }
    for (int t = 0; t < T_FEAT; ++t) {
        float nv = nr[t];
        s_u += nv * we[H_DIM + t];                    // wu_t = we[H:H+T]
        s_v += nv * we[2 * H_DIM + T_FEAT + t];       // wv_t = we[2H+T:2H+2T]
        const float* wtu = Wt + (size_t)(H_DIM + t) * R_REL;
        const float* wtv = Wt + (size_t)(2 * H_DIM + T_FEAT + t) * R_REL;
#pragma unroll
        for (int r = 0; r < R_REL; ++r) { au[r] += nv * wtu[r]; av[r] += nv * wtv[r]; }
    }
    su[n] = s_u; sv[n] = s_v;
#pragma unroll
    for (int r = 0; r < R_REL; ++r) {
        Au[(size_t)n * R_REL + r] = au[r];
        Av[(size_t)n * R_REL + r] = av[r];
    }
}

// Per-batch scalars: sg[b] = gz@wg + be, Ag[b,:] = gz@Wt[2H+2T:] + bt
__global__ void batch_scores(const float* __restrict__ gz, const float* __restrict__ we,
                             const float* __restrict__ be_p, const float* __restrict__ Wt,
                             const float* __restrict__ bt,
                             float* __restrict__ sg, float* __restrict__ Ag)
{
    int b = threadIdx.x;
    if (b >= BATCH) return;
    const float* g = gz + (size_t)b * H_DIM;
    float s = be_p[0];
    float ag[R_REL] = {bt[0], bt[1], bt[2], bt[3]};
    for (int k = 0; k < H_DIM; ++k) {
        float gv = g[k];
        s += gv * we[2 * H_DIM + 2 * T_FEAT + k];
        const float* w = Wt + (size_t)(2 * H_DIM + 2 * T_FEAT + k) * R_REL;
#pragma unroll
        for (int r = 0; r < R_REL; ++r) ag[r] += gv * w[r];
    }
    sg[b] = s;
#pragma unroll
    for (int r = 0; r < R_REL; ++r) Ag[(size_t)b * R_REL + r] = ag[r];
}

// pair_out (B,NG,NG,5): [sigmoid(su+sv+sg), Au+Av+Ag (4)]; float4 coalesced stores
__global__ void __launch_bounds__(256)
pair_assemble(const float* __restrict__ su, const float* __restrict__ sv,
              const float* __restrict__ sg, const float* __restrict__ Au,
              const float* __restrict__ Av, const float* __restrict__ Ag,
              float* __restrict__ outp)
{
    const long long total = (long long)BATCH * NGRP * NGRP * 5;
    long long base = ((long long)blockIdx.x * blockDim.x + threadIdx.x) * 4;
    if (base >= total) return;
    float4 res;
    float* rp = &res.x;
#pragma unroll
    for (int j = 0; j < 4; ++j) {
        long long e = base + j;
        int c = (int)(e % 5);
        long long p = e / 5;
        int v = (int)(p % NGRP);
        long long q = p / NGRP;
        int u = (int)(q % NGRP);
        int b = (int)(q / NGRP);
        int nu = b * NGRP + u;
        int nv = b * NGRP + v;
        float val;
        if (c == 0) {
            float x = su[nu] + sv[nv] + sg[b];
            val = 1.0f / (1.0f + expf(-x));
        } else {
            int r = c - 1;
            val = Au[(size_t)nu * R_REL + r] + Av[(size_t)nv * R_REL + r]
                + Ag[b * R_REL + r];
        }
        rp[j] = val;
    }
    *(float4*)(outp + base) = res;
}

// ---------------------------------------------------------------------------
extern "C" void kernel_launch(void* const* d_in, const int* in_sizes, int n_in,
                              void* d_out, int out_size, void* d_ws, size_t ws_size,
                              hipStream_t stream)
{
    const float* n_feats = (const float*)d_in[0];
    const float* eps     = (const float*)d_in[1];
    const int*   esrc    = (const int*)d_in[2];
    const int*   edst    = (const int*)d_in[3];
    const int*   ety     = (const int*)d_in[4];
    const float* W1   = (const float*)d_in[5];
    const float* Ws1  = (const float*)d_in[6];
    const float* b1   = (const float*)d_in[7];
    const float* W2   = (const float*)d_in[8];
    const float* Ws2  = (const float*)d_in[9];
    const float* b2   = (const float*)d_in[10];
    const float* Wmu  = (const float*)d_in[11];
    const float* Wsmu = (const float*)d_in[12];
    const float* bmu  = (const float*)d_in[13];
    const float* Wls  = (const float*)d_in[14];
    const float* Wsls = (const float*)d_in[15];
    const float* bls  = (const float*)d_in[16];
    const float* Wnt  = (const float*)d_in[17];
    const float* bnt  = (const float*)d_in[18];
    const float* we   = (const float*)d_in[19];
    const float* be   = (const float*)d_in[20];
    const float* Wt   = (const float*)d_in[21];
    const float* bt   = (const float*)d_in[22];

    float* out = (float*)d_out;
    const size_t PAIR = (size_t)BATCH * NGRP * NGRP * 5;
    float* nt_out = out + PAIR;                             // (N,T)
    float* mu_out = nt_out + (size_t)N_NODES * T_FEAT;      // (N,H)
    float* ls_out = mu_out + (size_t)N_NODES * H_DIM;       // (N,H)

    float* ws = (float*)d_ws;
    const size_t NH = (size_t)N_NODES * H_DIM;
    float* h1      = ws;
    float* h2      = ws + NH;
    float* tmp     = ws + 2 * NH;
    float* zb      = ws + 3 * NH;
    float* su      = ws + 4 * NH;
    float* sv      = su + N_NODES;
    float* Au      = sv + N_NODES;
    float* Av      = Au + (size_t)N_NODES * R_REL;
    float* gz      = Av + (size_t)N_NODES * R_REL;
    float* sg      = gz + BATCH * H_DIM;
    float* Ag      = sg + BATCH;
    float* nf_pad  = Ag + BATCH * R_REL;                    // N x 16
    float* w1_pad  = nf_pad + (size_t)N_NODES * K_PAD;      // R x 16 x 128
    float* ws1_pad = w1_pad + (size_t)R_REL * K_PAD * H_DIM;// 16 x 128

    dim3 gblk(256), ggrd(N_NODES / 16);
    const int sgrid = (E_EDGES * 32 + 255) / 256;

    // ---- pad layer-1 operands to K=16 (guard-free WMMA loops) ----
    pad_rows16<<<(N_NODES * K_PAD + 255) / 256, 256, 0, stream>>>(n_feats, nf_pad,
                                                                  N_NODES, T_FEAT);
    pad_w16<<<(R_REL * K_PAD * H_DIM + 255) / 256, 256, 0, stream>>>(W1, w1_pad, R_REL);
    pad_w16<<<(K_PAD * H_DIM + 255) / 256, 256, 0, stream>>>(Ws1, ws1_pad, 1);

    auto launch_gemm = [&](const float* Ain, int K, const float* Bmw,
                           const float* bias, int hb, float* Cout) {
        if (K == K_PAD)
            gemm_n128_wmma<K_PAD><<<ggrd, gblk, 0, stream>>>(Ain, Bmw, bias, hb, Cout);
        else
            gemm_n128_wmma<H_DIM><<<ggrd, gblk, 0, stream>>>(Ain, Bmw, bias, hb, Cout);
    };

    auto relconv = [&](const float* hin, int K, const float* Wr, const float* Wself,
                       const float* bias, float* outp, bool dorelu) {
        // self term (also initializes outp with bias)
        launch_gemm(hin, K, Wself, bias, 1, outp);
        for (int r = 0; r < R_REL; ++r) {
            launch_gemm(hin, K, Wr + (size_t)r * K * H_DIM, nullptr, 0, tmp);
            scatter_add_rel<<<sgrid, 256, 0, stream>>>(tmp, outp, esrc, edst, ety, r);
        }
        if (dorelu)
            relu_inplace<<<(int)((NH + 255) / 256), 256, 0, stream>>>(outp, (int)NH);
    };

    relconv(nf_pad, K_PAD, w1_pad, ws1_pad, b1, h1, true);
    relconv(h1, H_DIM, W2,  Ws2,  b2,  h2,     true);
    relconv(h2, H_DIM, Wmu, Wsmu, bmu, mu_out, false);
    relconv(h2, H_DIM, Wls, Wsls, bls, ls_out, false);

    z_combine<<<(int)((NH + 255) / 256), 256, 0, stream>>>(mu_out, ls_out, eps, zb, (int)NH);
    nt_gemv<<<(N_NODES * T_FEAT + 255) / 256, 256, 0, stream>>>(zb, Wnt, bnt, nt_out);
    gz_mean<<<(BATCH * H_DIM + 255) / 256, 256, 0, stream>>>(zb, gz);
    node_scores<<<N_NODES / 256, 256, 0, stream>>>(zb, nt_out, we, Wt, su, sv, Au, Av);
    batch_scores<<<1, 32, 0, stream>>>(gz, we, be, Wt, bt, sg, Ag);

    const long long total4 = (long long)PAIR / 4;
    pair_assemble<<<(int)((total4 + 255) / 256), 256, 0, stream>>>(su, sv, sg, Au, Av, Ag, out);
}